// GraphSAGE2_69286412419426
// MI455X (gfx1250) — compile-verified
//
#include <hip/hip_runtime.h>

#define N_NODES 50000
#define N_EDGES 600000
#define D_IN    128
#define D_HID   256
#define D_EMB   128
#define D_OUT   2

typedef __attribute__((ext_vector_type(2))) float v2f;
typedef __attribute__((ext_vector_type(8))) float v8f;

// ---------------------------------------------------------------------------
// Zero a float buffer (vectorized, grid-stride). n must be divisible by 4.
// ---------------------------------------------------------------------------
__global__ void zero_f32x4(float* __restrict__ p, int n4) {
    int i = blockIdx.x * blockDim.x + threadIdx.x;
    int stride = gridDim.x * blockDim.x;
    float4 z = make_float4(0.f, 0.f, 0.f, 0.f);
    float4* p4 = reinterpret_cast<float4*>(p);
    for (; i < n4; i += stride) p4[i] = z;
}

// ---------------------------------------------------------------------------
// In-degree counts (depends only on dst -> computed once, reused all layers)
// ---------------------------------------------------------------------------
__global__ void degree_kernel(const int* __restrict__ dst, float* __restrict__ cnt) {
    int e = blockIdx.x * blockDim.x + threadIdx.x;
    if (e < N_EDGES) atomicAdd(&cnt[dst[e]], 1.0f);
}

// ---------------------------------------------------------------------------
// Edge scatter-add: agg[dst] += x[src].  One thread per (edge, float4 chunk).
// ---------------------------------------------------------------------------
__global__ void scatter_kernel(const float* __restrict__ x,
                               const int* __restrict__ src,
                               const int* __restrict__ dst,
                               float* __restrict__ agg,
                               int K, int cshift, int total) {
    int tid = blockIdx.x * blockDim.x + threadIdx.x;
    if (tid >= total) return;
    int e = tid >> cshift;
    int c = tid & ((1 << cshift) - 1);
    int s = src[e];
    int d = dst[e];
    const float4 v = reinterpret_cast<const float4*>(x + (long)s * K)[c];
    float* a = agg + (long)d * K + c * 4;
    atomicAdd(a + 0, v.x);
    atomicAdd(a + 1, v.y);
    atomicAdd(a + 2, v.z);
    atomicAdd(a + 3, v.w);
}

// ---------------------------------------------------------------------------
// Repack weights into WMMA-native layout:
//   Wp[g][c][q] = W[4g+q][c]   (g = K/4 group, c = column, q = 0..3)
// A lane's per-step B pair (K=4g+koff, 4g+koff+1 at column c) becomes one
// contiguous 8-byte load; a wave's per-group footprint is one 256B span.
// ---------------------------------------------------------------------------
__global__ void repack_kernel(const float* __restrict__ W, float* __restrict__ Wp,
                              int K, int MOUT) {
    int t = blockIdx.x * blockDim.x + threadIdx.x;   // one per (g, c)
    int total = (K / 4) * MOUT;
    if (t >= total) return;
    int g = t / MOUT;
    int c = t % MOUT;
    float4 v;
    v.x = W[(long)(4 * g + 0) * MOUT + c];
    v.y = W[(long)(4 * g + 1) * MOUT + c];
    v.z = W[(long)(4 * g + 2) * MOUT + c];
    v.w = W[(long)(4 * g + 3) * MOUT + c];
    reinterpret_cast<float4*>(Wp)[t] = v;
}

// ---------------------------------------------------------------------------
// Fused SAGE layer GEMM:
//   out[tile] = act( (agg/deg) @ Wl + x @ Wr + b )
// Block = 256 threads (8 wave32s), SUB*16 rows.  A-tiles (mean + x) staged in
// LDS with degree normalization fused in.  Each wave owns one output column
// tile at a time and SUB independent 16x16 accumulator chains; packed-B
// operands are software-pipelined through double-buffered register chunks
// (one b64 load per WMMA K-step) so global loads overlap WMMA execution.
// ---------------------------------------------------------------------------
template <int K, int MOUT, bool RELU, int SUB>
__global__ __launch_bounds__(256) void sage_gemm(
    const float* __restrict__ agg,   // [N, K]  unnormalized neighbor sums
    const float* __restrict__ xin,   // [N, K]  self features
    const float* __restrict__ cnt,   // [N]     in-degrees
    const float* __restrict__ Wlp,   // packed [K/4, MOUT, 4]
    const float* __restrict__ Wrp,   // packed [K/4, MOUT, 4]
    const float* __restrict__ bias,  // [MOUT]
    float* __restrict__ out,         // [N, MOUT]
    int rowStart)
{
    constexpr int ROWS = SUB * 16;
    __shared__ float lmean[ROWS][K];
    __shared__ float lx[ROWS][K];

    const int tid = threadIdx.x;
    const int rowBase = rowStart + blockIdx.x * ROWS;

    // ---- stage A-tiles (mean normalized on the fly) ----
    constexpr int C4 = K / 4;
    constexpr int N4 = ROWS * C4;
    for (int i = tid; i < N4; i += 256) {
        const int r  = i / C4;
        const int c4 = i % C4;
        const long rowOff = (long)(rowBase + r) * K;
        float4 a = reinterpret_cast<const float4*>(agg + rowOff)[c4];
        const float rcp = 1.0f / fmaxf(cnt[rowBase + r], 1.0f);
        a.x *= rcp; a.y *= rcp; a.z *= rcp; a.w *= rcp;
        reinterpret_cast<float4*>(&lmean[r][0])[c4] = a;
        reinterpret_cast<float4*>(&lx[r][0])[c4] =
            reinterpret_cast<const float4*>(xin + rowOff)[c4];
    }
    __syncthreads();

    const int lane    = tid & 31;
    const int wave    = tid >> 5;
    const int halfSel = lane >> 4;       // 0: lanes 0-15, 1: lanes 16-31
    const int lrow    = lane & 15;
    const int koff    = halfSel * 2;     // A/B K-offset within each group of 4

    constexpr int STEPS = K / 4;         // WMMA K-steps per pass
    constexpr int CH    = 8;             // steps per prefetch chunk
    constexpr int NCH   = STEPS / CH;

    for (int ct = wave; ct < MOUT / 16; ct += 8) {
        const int col = ct * 16 + lrow;
        v8f acc[SUB];
        #pragma unroll
        for (int s = 0; s < SUB; ++s) acc[s] = (v8f){};

        #pragma unroll
        for (int pass = 0; pass < 2; ++pass) {
            const float* __restrict__ Wp = pass ? Wrp : Wlp;
            const float (*tile)[K] = pass ? lx : lmean;
            // lane's packed-B base: group g lives at Wp + (g*MOUT + col)*4 + koff
            const float* __restrict__ wlane = Wp + (long)col * 4 + koff;

            v2f bb[2][CH];
            // preload chunk 0
            #pragma unroll
            for (int j = 0; j < CH; ++j)
                bb[0][j] = *reinterpret_cast<const v2f*>(wlane + (long)j * (MOUT * 4));
            #pragma unroll
            for (int kc = 0; kc < NCH; ++kc) {
                const int cur = kc & 1;
                if (kc + 1 < NCH) {
                    // issue next chunk's loads before consuming this chunk
                    #pragma unroll
                    for (int j = 0; j < CH; ++j) {
                        const long g = CH * (kc + 1) + j;
                        bb[cur ^ 1][j] =
                            *reinterpret_cast<const v2f*>(wlane + g * (MOUT * 4));
                    }
                }
                #pragma unroll
                for (int j = 0; j < CH; ++j) {
                    const int k = 4 * (CH * kc + j) + koff;
                    #pragma unroll
                    for (int s = 0; s < SUB; ++s) {   // independent acc chains
                        v2f a;
                        a.x = tile[s * 16 + lrow][k];
                        a.y = tile[s * 16 + lrow][k + 1];
                        acc[s] = __builtin_amdgcn_wmma_f32_16x16x4_f32(
                            false, a, false, bb[cur][j], (short)0, acc[s],
                            false, false);
                    }
                }
            }
        }

        const float bv = bias[col];
        #pragma unroll
        for (int s = 0; s < SUB; ++s) {
            #pragma unroll
            for (int i = 0; i < 8; ++i) {
                const int r = rowBase + s * 16 + i + halfSel * 8;  // C/D layout
                float v = acc[s][i] + bv;
                if (RELU) v = fmaxf(v, 0.0f);
                out[(long)r * MOUT + col] = v;
            }
        }
    }
}

// ---------------------------------------------------------------------------
// Output head: out = h3 @ Wout + bout  (MOUT=2, too narrow for WMMA)
// ---------------------------------------------------------------------------
__global__ void head_kernel(const float* __restrict__ h,
                            const float* __restrict__ Wout,  // [D_EMB, 2]
                            const float* __restrict__ bout,  // [2]
                            float* __restrict__ out) {       // [N, 2]
    int n = blockIdx.x * blockDim.x + threadIdx.x;
    if (n >= N_NODES) return;
    float a0 = bout[0], a1 = bout[1];
    const float* hp = h + (long)n * D_EMB;
    #pragma unroll 4
    for (int k = 0; k < D_EMB; ++k) {
        const float v = hp[k];
        a0 += v * Wout[k * 2 + 0];
        a1 += v * Wout[k * 2 + 1];
    }
    out[n * 2 + 0] = a0;
    out[n * 2 + 1] = a1;
}

// ---------------------------------------------------------------------------
template <int K, int MOUT, bool RELU>
static void launch_sage_gemm(const float* agg, const float* xin, const float* cnt,
                             const float* Wlp, const float* Wrp, const float* bias,
                             float* out, hipStream_t stream) {
    // 50000 = 32*1562 + 16 : full 32-row blocks + one 16-row remainder block
    constexpr int FULL = N_NODES / 32;           // 1562
    sage_gemm<K, MOUT, RELU, 2><<<FULL, 256, 0, stream>>>(
        agg, xin, cnt, Wlp, Wrp, bias, out, 0);
    sage_gemm<K, MOUT, RELU, 1><<<1, 256, 0, stream>>>(
        agg, xin, cnt, Wlp, Wrp, bias, out, FULL * 32);
}

extern "C" void kernel_launch(void* const* d_in, const int* in_sizes, int n_in,
                              void* d_out, int out_size, void* d_ws, size_t ws_size,
                              hipStream_t stream) {
    const float* x    = (const float*)d_in[0];
    const int*   ei   = (const int*)d_in[1];          // [2, E] row-major (int32)
    const float* Wl1  = (const float*)d_in[2];
    const float* Wr1  = (const float*)d_in[3];
    const float* b1   = (const float*)d_in[4];
    const float* Wl2  = (const float*)d_in[5];
    const float* Wr2  = (const float*)d_in[6];
    const float* b2   = (const float*)d_in[7];
    const float* Wl3  = (const float*)d_in[8];
    const float* Wr3  = (const float*)d_in[9];
    const float* b3   = (const float*)d_in[10];
    const float* Wout = (const float*)d_in[11];
    const float* bout = (const float*)d_in[12];
    float* out = (float*)d_out;

    const int* src = ei;
    const int* dst = ei + N_EDGES;

    // Workspace layout (floats):
    //   cnt[N] | agg[N*256] | h1[N*256] | h2[N*256] | packed weights (1 MB)
    float* ws   = (float*)d_ws;
    float* cnt  = ws;
    float* agg  = cnt + ((N_NODES + 63) / 64) * 64;
    float* h1   = agg + (long)N_NODES * 256;
    float* h2   = h1 + (long)N_NODES * 256;
    float* h3   = h1;   // h1 dead after layer-2 GEMM; reuse for h3
    float* wl1p = h2 + (long)N_NODES * 256;
    float* wr1p = wl1p + D_IN * D_HID;    // 32768
    float* wl2p = wr1p + D_IN * D_HID;    // 32768
    float* wr2p = wl2p + D_HID * D_HID;   // 65536
    float* wl3p = wr2p + D_HID * D_HID;   // 65536
    float* wr3p = wl3p + D_HID * D_EMB;   // 32768

    const int T = 256;

    // ---- repack weights (once per launch; L2-resident afterwards) ----
    repack_kernel<<<(D_IN  / 4 * D_HID + T - 1) / T, T, 0, stream>>>(Wl1, wl1p, D_IN,  D_HID);
    repack_kernel<<<(D_IN  / 4 * D_HID + T - 1) / T, T, 0, stream>>>(Wr1, wr1p, D_IN,  D_HID);
    repack_kernel<<<(D_HID / 4 * D_HID + T - 1) / T, T, 0, stream>>>(Wl2, wl2p, D_HID, D_HID);
    repack_kernel<<<(D_HID / 4 * D_HID + T - 1) / T, T, 0, stream>>>(Wr2, wr2p, D_HID, D_HID);
    repack_kernel<<<(D_HID / 4 * D_EMB + T - 1) / T, T, 0, stream>>>(Wl3, wl3p, D_HID, D_EMB);
    repack_kernel<<<(D_HID / 4 * D_EMB + T - 1) / T, T, 0, stream>>>(Wr3, wr3p, D_HID, D_EMB);

    // ---- degrees (once) ----
    zero_f32x4<<<(N_NODES / 4 + T - 1) / T, T, 0, stream>>>(cnt, N_NODES / 4);
    degree_kernel<<<(N_EDGES + T - 1) / T, T, 0, stream>>>(dst, cnt);

    // ---- layer 1: K=128 -> 256, ReLU ----
    {
        const int n4 = N_NODES * D_IN / 4;
        zero_f32x4<<<4096, T, 0, stream>>>(agg, n4);
        const int total = N_EDGES * (D_IN / 4);           // cshift = 5
        scatter_kernel<<<(total + T - 1) / T, T, 0, stream>>>(x, src, dst, agg,
                                                              D_IN, 5, total);
        launch_sage_gemm<128, 256, true>(agg, x, cnt, wl1p, wr1p, b1, h1, stream);
    }
    // ---- layer 2: K=256 -> 256, ReLU ----
    {
        const int n4 = N_NODES * D_HID / 4;
        zero_f32x4<<<4096, T, 0, stream>>>(agg, n4);
        const int total = N_EDGES * (D_HID / 4);          // cshift = 6
        scatter_kernel<<<(total + T - 1) / T, T, 0, stream>>>(h1, src, dst, agg,
                                                              D_HID, 6, total);
        launch_sage_gemm<256, 256, true>(agg, h1, cnt, wl2p, wr2p, b2, h2, stream);
    }
    // ---- layer 3: K=256 -> 128, no activation ----
    {
        const int n4 = N_NODES * D_HID / 4;
        zero_f32x4<<<4096, T, 0, stream>>>(agg, n4);
        const int total = N_EDGES * (D_HID / 4);          // cshift = 6
        scatter_kernel<<<(total + T - 1) / T, T, 0, stream>>>(h2, src, dst, agg,
                                                              D_HID, 6, total);
        launch_sage_gemm<256, 128, false>(agg, h2, cnt, wl3p, wr3p, b3, h3, stream);
    }
    // ---- output head ----
    head_kernel<<<(N_NODES + T - 1) / T, T, 0, stream>>>(h3, Wout, bout, out);
}